// GCN_10471130267748
// MI455X (gfx1250) — compile-verified
//
#include <hip/hip_runtime.h>
#include <math.h>

typedef float v2f __attribute__((ext_vector_type(2)));
typedef float v8f __attribute__((ext_vector_type(8)));
typedef unsigned int u32x4 __attribute__((ext_vector_type(4)));
typedef int i32x4 __attribute__((ext_vector_type(4)));
typedef int i32x8 __attribute__((ext_vector_type(8)));

#define FIN  512
#define HID  16
#define NCLS 40

static __device__ __forceinline__ v8f wmma_f32_16x16x4(v2f a, v2f b, v8f c) {
  // 8 args: (neg_a, A, neg_b, B, c_mod, C, reuse_a, reuse_b)
  return __builtin_amdgcn_wmma_f32_16x16x4_f32(false, a, false, b, (short)0, c, false, false);
}

// TDM: DMA `nelem` contiguous f32 from global into LDS (one descriptor, one wave issues).
// D# layout per CDNA5 ISA ch.8: group0 = {flags/count, lds_addr, global_addr, type=2},
// group1 = {data_size=4B, tensor_dim0=tile_dim0=nelem, dim1=1, stride0=nelem}.
static __device__ __forceinline__ void tdm_load_f32_to_lds(const float* gsrc,
                                                           void* lds_dst,
                                                           unsigned nelem) {
  unsigned lds_off = (unsigned)(unsigned long long)(uintptr_t)lds_dst; // generic LDS addr[31:0] = byte offset
  unsigned long long ga = (unsigned long long)(uintptr_t)gsrc;
  u32x4 g0;
  g0.x = 1u;                                   // count=1 (valid), user mode, gather off
  g0.y = lds_off;                              // D#.lds_addr
  g0.z = (unsigned)ga;                         // global_addr[31:0]   (bits 95:64)
  g0.w = (unsigned)(ga >> 32) | 0x80000000u;   // global_addr[56:32] | type=2 (bits 127:126)
  i32x8 g1;
  g1[0] = (int)(2u << 16);                     // workgroup_mask=0, data_size=2 -> 4 bytes
  g1[1] = (int)((nelem & 0xffffu) << 16);      // tensor_dim0[15:0] (bits 63:48)
  g1[2] = (int)((nelem >> 16) | (1u << 16));   // tensor_dim0[31:16], tensor_dim1=1
  g1[3] = (int)((nelem & 0xffffu) << 16);      // tile_dim0 (bits 127:112)
  g1[4] = 1;                                   // tile_dim1=1, tile_dim2=0
  g1[5] = (int)nelem;                          // tensor_dim0_stride[31:0]
  g1[6] = 0;
  g1[7] = 0;
  i32x4 z4 = {0, 0, 0, 0};                     // groups 2/3 unused (<=2D tensor)
  i32x8 z8 = {0, 0, 0, 0, 0, 0, 0, 0};
  __builtin_amdgcn_tensor_load_to_lds(g0, g1, z4, z4, z8, 0);
}

// ---------------- norm pipeline ----------------

__global__ __launch_bounds__(256) void init_kernel(float* deg, float* agg1, float* agg2, int n) {
  long long i = (long long)blockIdx.x * blockDim.x + threadIdx.x;
  if (i < n) deg[i] = 1.0f;                 // self-loop weight pre-added
  if (i < (long long)n * HID)  agg1[i] = 0.0f;
  if (i < (long long)n * NCLS) agg2[i] = 0.0f;
}

__global__ __launch_bounds__(256) void deg_kernel(const int* __restrict__ col,
                                                  const float* __restrict__ w,
                                                  float* __restrict__ deg, int e) {
  int i = blockIdx.x * blockDim.x + threadIdx.x;
  if (i >= e) return;
  atomicAdd(deg + col[i], w[i]);
}

__global__ __launch_bounds__(256) void dinv_kernel(const float* __restrict__ deg,
                                                   float* __restrict__ dinv, int n) {
  int i = blockIdx.x * blockDim.x + threadIdx.x;
  if (i >= n) return;
  float d = deg[i];
  dinv[i] = (d > 0.0f) ? rsqrtf(d) : 0.0f;
}

__global__ __launch_bounds__(256) void norm_kernel(const int* __restrict__ row,
                                                   const int* __restrict__ col,
                                                   const float* __restrict__ w,
                                                   const float* __restrict__ dinv,
                                                   float* __restrict__ nrm, int e) {
  int i = blockIdx.x * blockDim.x + threadIdx.x;
  if (i >= e) return;
  nrm[i] = dinv[row[i]] * w[i] * dinv[col[i]];
}

// ---------------- GEMM1: t1 = x @ W1  (WMMA f32 16x16x4, W1 via TDM) ----------------

__global__ __launch_bounds__(256) void gemm1_kernel(const float* __restrict__ x,
                                                    const float* __restrict__ W1,
                                                    float* __restrict__ t1, int n) {
  __shared__ float ldsW[FIN * HID];       // 32 KB of the 320 KB WGP LDS
  const int tid = threadIdx.x;
  if (tid < 32) {                         // wave 0 issues one TDM descriptor for all 32 KB
    tdm_load_f32_to_lds(W1, ldsW, FIN * HID);
    __builtin_amdgcn_s_wait_tensorcnt(0);
  }
  __syncthreads();

  const int wave = tid >> 5;
  const int lane = tid & 31;
  const int hl   = lane >> 4;             // 0: K pair {0,1}, 1: K pair {2,3}
  const int l15  = lane & 15;
  const int tile = blockIdx.x * 8 + wave;
  if (tile * 16 >= n) return;             // wave-uniform exit: EXEC stays all-1 for WMMA

  const int row = tile * 16 + l15;        // A: row M = lane&15
  const float2* __restrict__ xrow = (const float2*)(x + (size_t)row * FIN + 2 * hl);
  const float*  __restrict__ brow = ldsW + (2 * hl) * HID + l15;

  v8f c = {};
#pragma unroll 8
  for (int k = 0; k < FIN; k += 4) {
    float2 af = xrow[k >> 1];             // x[row, k+2*hl .. k+2*hl+1]
    v2f a; a.x = af.x; a.y = af.y;
    v2f b;                                // W1[k+2*hl, n], W1[k+2*hl+1, n]
    b.x = brow[k * HID];
    b.y = brow[k * HID + HID];
    c = wmma_f32_16x16x4(a, b, c);
  }

  float* __restrict__ out = t1 + (size_t)(tile * 16 + 8 * hl) * HID + l15;
#pragma unroll
  for (int v = 0; v < 8; ++v) out[(size_t)v * HID] = c[v];
}

// ---------------- edge scatter layer 1 (16 feat) ----------------

__global__ __launch_bounds__(256) void scatter1_kernel(const int* __restrict__ row,
                                                       const int* __restrict__ col,
                                                       const float* __restrict__ nrm,
                                                       const float* __restrict__ t1,
                                                       float* __restrict__ agg1, int e) {
  int i = blockIdx.x * blockDim.x + threadIdx.x;
  if (i >= e) return;
  const int r = row[i], cg = col[i];
  const float nw = nrm[i];
  const float4* __restrict__ src = (const float4*)(t1 + (size_t)r * HID);
  float* __restrict__ dst = agg1 + (size_t)cg * HID;
#pragma unroll
  for (int q = 0; q < HID / 4; ++q) {
    float4 v = src[q];
    atomicAdd(dst + q * 4 + 0, v.x * nw);
    atomicAdd(dst + q * 4 + 1, v.y * nw);
    atomicAdd(dst + q * 4 + 2, v.z * nw);
    atomicAdd(dst + q * 4 + 3, v.w * nw);
  }
}

// ---------------- bias + self-loop + ReLU ----------------

__global__ __launch_bounds__(256) void relu_kernel(const float* __restrict__ agg1,
                                                   const float* __restrict__ t1,
                                                   const float* __restrict__ dinv,
                                                   const float* __restrict__ b1,
                                                   float* __restrict__ h1, int n) {
  long long i = (long long)blockIdx.x * blockDim.x + threadIdx.x;
  if (i >= (long long)n * HID) return;
  int node = (int)(i >> 4), f = (int)(i & 15);
  float di = dinv[node];
  float v = agg1[i] + t1[i] * di * di + b1[f];
  h1[i] = fmaxf(v, 0.0f);
}

// ---------------- GEMM2: t2 = h1 @ W2  (WMMA f32, N=40 -> 3 tiles, W2 via TDM) ----------------

__global__ __launch_bounds__(256) void gemm2_kernel(const float* __restrict__ h1,
                                                    const float* __restrict__ W2,
                                                    float* __restrict__ t2, int n) {
  __shared__ float ldsW[HID * NCLS];      // 640 floats
  const int tid = threadIdx.x;
  if (tid < 32) {
    tdm_load_f32_to_lds(W2, ldsW, HID * NCLS);
    __builtin_amdgcn_s_wait_tensorcnt(0);
  }
  __syncthreads();

  const int wave = tid >> 5;
  const int lane = tid & 31;
  const int hl   = lane >> 4;
  const int l15  = lane & 15;
  const int tile = blockIdx.x * 8 + wave;
  if (tile * 16 >= n) return;

  const int row = tile * 16 + l15;
  const float2* __restrict__ arow = (const float2*)(h1 + (size_t)row * HID + 2 * hl);

  const int n2 = 32 + l15;
  const float mask2 = (n2 < NCLS) ? 1.0f : 0.0f;
  const int n2c = (n2 < NCLS) ? n2 : (NCLS - 1);

  v8f c0 = {}, c1 = {}, c2 = {};
#pragma unroll
  for (int kk = 0; kk < 4; ++kk) {
    float2 af = arow[kk * 2];             // h1[row, kk*4 + 2*hl .. +1]
    v2f a; a.x = af.x; a.y = af.y;
    const int kb = kk * 4 + 2 * hl;
    v2f b0, b1v, b2v;
    b0.x  = ldsW[kb * NCLS + l15];        b0.y  = ldsW[(kb + 1) * NCLS + l15];
    b1v.x = ldsW[kb * NCLS + 16 + l15];   b1v.y = ldsW[(kb + 1) * NCLS + 16 + l15];
    b2v.x = ldsW[kb * NCLS + n2c] * mask2;
    b2v.y = ldsW[(kb + 1) * NCLS + n2c] * mask2;
    c0 = wmma_f32_16x16x4(a, b0,  c0);    // A fragment reused across 3 column tiles
    c1 = wmma_f32_16x16x4(a, b1v, c1);
    c2 = wmma_f32_16x16x4(a, b2v, c2);
  }

  float* __restrict__ orow = t2 + (size_t)(tile * 16 + 8 * hl) * NCLS;
#pragma unroll
  for (int v = 0; v < 8; ++v) {
    size_t ro = (size_t)v * NCLS;
    orow[ro + l15]      = c0[v];
    orow[ro + 16 + l15] = c1[v];
    if (l15 < 8) orow[ro + 32 + l15] = c2[v];
  }
}

// ---------------- edge scatter layer 2 (40 feat) ----------------

__global__ __launch_bounds__(256) void scatter2_kernel(const int* __restrict__ row,
                                                       const int* __restrict__ col,
                                                       const float* __restrict__ nrm,
                                                       const float* __restrict__ t2,
                                                       float* __restrict__ agg2, int e) {
  int i = blockIdx.x * blockDim.x + threadIdx.x;
  if (i >= e) return;
  const int r = row[i], cg = col[i];
  const float nw = nrm[i];
  const float4* __restrict__ src = (const float4*)(t2 + (size_t)r * NCLS);
  float* __restrict__ dst = agg2 + (size_t)cg * NCLS;
#pragma unroll
  for (int q = 0; q < NCLS / 4; ++q) {
    float4 v = src[q];
    atomicAdd(dst + q * 4 + 0, v.x * nw);
    atomicAdd(dst + q * 4 + 1, v.y * nw);
    atomicAdd(dst + q * 4 + 2, v.z * nw);
    atomicAdd(dst + q * 4 + 3, v.w * nw);
  }
}

// ---------------- bias + self-loop + log_softmax ----------------

__global__ __launch_bounds__(256) void lsm_kernel(const float* __restrict__ agg2,
                                                  const float* __restrict__ t2,
                                                  const float* __restrict__ dinv,
                                                  const float* __restrict__ b2,
                                                  float* __restrict__ out, int n) {
  int i = blockIdx.x * blockDim.x + threadIdx.x;
  if (i >= n) return;
  float di = dinv[i];
  float s = di * di;
  float v[NCLS];
  float mx = -INFINITY;
  const float* __restrict__ ar = agg2 + (size_t)i * NCLS;
  const float* __restrict__ tr = t2 + (size_t)i * NCLS;
#pragma unroll
  for (int c = 0; c < NCLS; ++c) {
    v[c] = ar[c] + tr[c] * s + b2[c];
    mx = fmaxf(mx, v[c]);
  }
  float sum = 0.0f;
#pragma unroll
  for (int c = 0; c < NCLS; ++c) sum += __expf(v[c] - mx);
  float lse = __logf(sum) + mx;
  float* __restrict__ orow = out + (size_t)i * NCLS;
#pragma unroll
  for (int c = 0; c < NCLS; ++c) orow[c] = v[c] - lse;
}

// ---------------- launch ----------------

extern "C" void kernel_launch(void* const* d_in, const int* in_sizes, int n_in,
                              void* d_out, int out_size, void* d_ws, size_t ws_size,
                              hipStream_t stream) {
  const float* x  = (const float*)d_in[0];
  const float* ew = (const float*)d_in[1];
  const float* W1 = (const float*)d_in[2];
  const float* b1 = (const float*)d_in[3];
  const float* W2 = (const float*)d_in[4];
  const float* b2 = (const float*)d_in[5];
  const int* eidx = (const int*)d_in[6];

  const int n = in_sizes[0] / FIN;    // 100000
  const int e = in_sizes[1];          // 3200000
  const int* row = eidx;
  const int* col = eidx + e;

  float* ws = (float*)d_ws;
  size_t off = 0;
  float* deg  = ws + off; off += n;
  float* dinv = ws + off; off += n;
  float* nrm  = ws + off; off += e;
  float* t1   = ws + off; off += (size_t)n * HID;
  float* agg1 = ws + off; off += (size_t)n * HID;
  float* h1   = ws + off; off += (size_t)n * HID;
  float* t2   = ws + off; off += (size_t)n * NCLS;
  float* agg2 = ws + off; off += (size_t)n * NCLS;
  float* outf = (float*)d_out;

  const int B = 256;
  const int gInit = (int)(((long long)n * NCLS + B - 1) / B);
  const int gE    = (e + B - 1) / B;
  const int gN    = (n + B - 1) / B;
  const int tiles = (n + 15) / 16;
  const int gGemm = (tiles + 7) / 8;          // 8 waves/block, 1 tile/wave
  const int gH    = (int)(((long long)n * HID + B - 1) / B);

  init_kernel<<<gInit, B, 0, stream>>>(deg, agg1, agg2, n);
  deg_kernel<<<gE, B, 0, stream>>>(col, ew, deg, e);
  dinv_kernel<<<gN, B, 0, stream>>>(deg, dinv, n);
  norm_kernel<<<gE, B, 0, stream>>>(row, col, ew, dinv, nrm, e);

  gemm1_kernel<<<gGemm, B, 0, stream>>>(x, W1, t1, n);
  scatter1_kernel<<<gE, B, 0, stream>>>(row, col, nrm, t1, agg1, e);
  relu_kernel<<<gH, B, 0, stream>>>(agg1, t1, dinv, b1, h1, n);

  gemm2_kernel<<<gGemm, B, 0, stream>>>(h1, W2, t2, n);
  scatter2_kernel<<<gE, B, 0, stream>>>(row, col, nrm, t2, agg2, e);
  lsm_kernel<<<gN, B, 0, stream>>>(agg2, t2, dinv, b2, outf, n);
}